// SAGE_70849780515474
// MI455X (gfx1250) — compile-verified
//
#include <hip/hip_runtime.h>
#include <hip/hip_bf16.h>
#include <math.h>

typedef __attribute__((ext_vector_type(2))) float v2f;
typedef __attribute__((ext_vector_type(8))) float v8f;

// ---------------------------------------------------------------------------
// Zero-fill scratch accumulators (graph-capture safe, deterministic).
// ---------------------------------------------------------------------------
__global__ void sage_zero(float* __restrict__ p, size_t n) {
    size_t i = (size_t)blockIdx.x * blockDim.x + threadIdx.x;
    size_t stride = (size_t)gridDim.x * blockDim.x;
    for (; i < n; i += stride) p[i] = 0.0f;
}

// ---------------------------------------------------------------------------
// GEMM1: y[N x 128] = x[N x 256] @ [W1_l | W1_r]  (fp32 WMMA 16x16x4)
// Block = 256 threads = 8 waves. Block owns 16 rows; wave w owns columns
// [w*16, w*16+16). A-tile (16x256) staged in LDS once, shared by all 8 waves.
// Row guards are uniform per block (clamped loads + one fullTile branch) so
// the hot path has no exec-mask juggling.
// ---------------------------------------------------------------------------
#define LDA 260  // 16B-aligned rows (260*4 % 16 == 0), bank-conflict free
__global__ __launch_bounds__(256)
void sage_gemm1(const float* __restrict__ x, const float* __restrict__ Wl,
                const float* __restrict__ Wr, float* __restrict__ y, int N) {
    __shared__ float lA[16 * LDA];
    const int t = threadIdx.x;
    const int mBase = blockIdx.x * 16;
    const bool fullTile = (mBase + 16 <= N);   // block-uniform

    // Coalesced float4 load of the 16x256 A tile (1024 float4 / 256 threads).
    // Clamp row index instead of branching: padding rows read valid memory,
    // their results are never stored.
    for (int j = 0; j < 4; ++j) {
        int chunk = t + j * 256;
        int row = chunk >> 6;            // chunk/64
        int col = (chunk & 63) << 2;     // 4-float granules
        int node = mBase + row;
        if (node >= N) node = N - 1;
        float4 v = *(const float4*)(x + (size_t)node * 256 + col);
        *(float4*)(&lA[row * LDA + col]) = v;
    }
    __syncthreads();

    const int w    = t >> 5;         // wave id -> 16-wide column tile
    const int lane = t & 31;
    const int lr   = lane & 15;      // A row / B column within tile
    const int hi   = lane >> 4;      // 0 -> K pair {0,1}; 1 -> K pair {2,3}
    const float* Wsel = (w < 4) ? Wl : Wr;     // cols 0-63 / 64-127
    const int wc = (w & 3) * 16 + lr;          // column inside 64-wide W

    v8f acc = {};
    for (int k0 = 0; k0 < 256; k0 += 4) {
        const int ka = k0 + hi * 2;
        v2f a, b;
        a.x = lA[lr * LDA + ka];
        a.y = lA[lr * LDA + ka + 1];
        b.x = Wsel[(size_t)ka * 64 + wc];
        b.y = Wsel[(size_t)(ka + 1) * 64 + wc];
        acc = __builtin_amdgcn_wmma_f32_16x16x4_f32(
            false, a, false, b, (short)0, acc, false, false);
    }

    // C/D layout: vgpr r, lanes 0-15 -> M=r, lanes 16-31 -> M=8+r; N=lane%16.
    const int col = w * 16 + lr;
    float* yrow = y + (size_t)(mBase + hi * 8) * 128 + col;
    if (fullTile) {
        #pragma unroll
        for (int r = 0; r < 8; ++r) yrow[(size_t)r * 128] = acc[r];
    } else {
        for (int r = 0; r < 8; ++r) {
            int node = mBase + hi * 8 + r;
            if (node < N) y[(size_t)node * 128 + col] = acc[r];
        }
    }
}

// ---------------------------------------------------------------------------
// Edge scatter 1: aggs[dst] += y_l[src] (64 floats/edge), deg[dst] += 1.
// ---------------------------------------------------------------------------
__global__ void sage_scatter1(const float* __restrict__ y, const int* __restrict__ src,
                              const int* __restrict__ dst, float* __restrict__ aggs,
                              float* __restrict__ deg, int E) {
    int t = blockIdx.x * blockDim.x + threadIdx.x;
    int e = t >> 6;
    int f = t & 63;
    if (e >= E) return;
    int s = src[e], d = dst[e];
    atomicAdd(&aggs[(size_t)d * 64 + f], y[(size_t)s * 128 + f]);
    if (f == 0) atomicAdd(&deg[d], 1.0f);
}

// ---------------------------------------------------------------------------
// Combine 1: h = relu(l2norm(aggs/max(deg,1) + b1 + y_r)). One wave per node.
// ---------------------------------------------------------------------------
__global__ __launch_bounds__(256)
void sage_combine1(const float* __restrict__ aggs, const float* __restrict__ deg,
                   const float* __restrict__ b1, const float* __restrict__ y,
                   float* __restrict__ h, int N) {
    int w = threadIdx.x >> 5, l = threadIdx.x & 31;
    int node = blockIdx.x * 8 + w;
    if (node >= N) return;
    float invd = 1.0f / fmaxf(deg[node], 1.0f);
    float v0 = aggs[(size_t)node * 64 + l]      * invd + b1[l]      + y[(size_t)node * 128 + 64 + l];
    float v1 = aggs[(size_t)node * 64 + 32 + l] * invd + b1[32 + l] + y[(size_t)node * 128 + 96 + l];
    float ss = v0 * v0 + v1 * v1;
    for (int off = 16; off > 0; off >>= 1) ss += __shfl_xor(ss, off, 32);
    float inv = 1.0f / fmaxf(sqrtf(ss), 1e-12f);
    h[(size_t)node * 64 + l]      = fmaxf(v0 * inv, 0.0f);
    h[(size_t)node * 64 + 32 + l] = fmaxf(v1 * inv, 0.0f);
}

// ---------------------------------------------------------------------------
// GEMM2: z[N x 8] = h[N x 64] @ [W2_l | W2_r] (cols 8..15 zero-padded).
// One wave per 16 nodes; 16 fp32 WMMA ops per wave. Clamped A loads keep the
// inner loop free of divergent exec masking.
// ---------------------------------------------------------------------------
__global__ __launch_bounds__(256)
void sage_gemm2(const float* __restrict__ h, const float* __restrict__ Wl,
                const float* __restrict__ Wr, float* __restrict__ z, int N) {
    const int w = threadIdx.x >> 5;
    const int lane = threadIdx.x & 31;
    const int mBase = (blockIdx.x * 8 + w) * 16;
    if (mBase >= N) return;                 // wave-uniform exit
    const bool fullTile = (mBase + 16 <= N);
    const int lr = lane & 15;
    const int hi = lane >> 4;
    int row = mBase + lr;
    if (row >= N) row = N - 1;              // clamp: padding rows never stored
    const float* hrow = h + (size_t)row * 64;

    v8f acc = {};
    for (int k0 = 0; k0 < 64; k0 += 4) {
        const int ka = k0 + hi * 2;
        v2f a, b = {0.f, 0.f};
        a.x = hrow[ka];
        a.y = hrow[ka + 1];
        if (lr < 4)      { b.x = Wl[ka * 4 + lr];       b.y = Wl[(ka + 1) * 4 + lr]; }
        else if (lr < 8) { b.x = Wr[ka * 4 + (lr - 4)]; b.y = Wr[(ka + 1) * 4 + (lr - 4)]; }
        acc = __builtin_amdgcn_wmma_f32_16x16x4_f32(
            false, a, false, b, (short)0, acc, false, false);
    }

    if (lr < 8) {
        float* zrow = z + (size_t)(mBase + hi * 8) * 8 + lr;
        if (fullTile) {
            #pragma unroll
            for (int r = 0; r < 8; ++r) zrow[(size_t)r * 8] = acc[r];
        } else {
            for (int r = 0; r < 8; ++r) {
                int node = mBase + hi * 8 + r;
                if (node < N) z[(size_t)node * 8 + lr] = acc[r];
            }
        }
    }
}

// ---------------------------------------------------------------------------
// Edge scatter 2: agg2[dst] += z_l[src] (4 floats/edge).
// ---------------------------------------------------------------------------
__global__ void sage_scatter2(const float* __restrict__ z, const int* __restrict__ src,
                              const int* __restrict__ dst, float* __restrict__ agg2, int E) {
    int t = blockIdx.x * blockDim.x + threadIdx.x;
    int e = t >> 2;
    int f = t & 3;
    if (e >= E) return;
    atomicAdd(&agg2[(size_t)dst[e] * 4 + f], z[(size_t)src[e] * 8 + f]);
}

// ---------------------------------------------------------------------------
// Combine 2: out = l2norm(agg2/max(deg,1) + b2 + z_r). One thread per node.
// ---------------------------------------------------------------------------
__global__ __launch_bounds__(256)
void sage_combine2(const float* __restrict__ agg2, const float* __restrict__ deg,
                   const float* __restrict__ b2, const float* __restrict__ z,
                   float* __restrict__ out, int N) {
    int node = blockIdx.x * blockDim.x + threadIdx.x;
    if (node >= N) return;
    float invd = 1.0f / fmaxf(deg[node], 1.0f);
    float v[4], ss = 0.0f;
    for (int c = 0; c < 4; ++c) {
        v[c] = agg2[(size_t)node * 4 + c] * invd + b2[c] + z[(size_t)node * 8 + 4 + c];
        ss += v[c] * v[c];
    }
    float inv = 1.0f / fmaxf(sqrtf(ss), 1e-12f);
    for (int c = 0; c < 4; ++c) out[(size_t)node * 4 + c] = v[c] * inv;
}

// ---------------------------------------------------------------------------
extern "C" void kernel_launch(void* const* d_in, const int* in_sizes, int n_in,
                              void* d_out, int out_size, void* d_ws, size_t ws_size,
                              hipStream_t stream) {
    const float* x   = (const float*)d_in[0];
    const int*   ei  = (const int*)d_in[1];
    const float* W1l = (const float*)d_in[2];
    const float* b1l = (const float*)d_in[3];
    const float* W1r = (const float*)d_in[4];
    const float* W2l = (const float*)d_in[5];
    const float* b2l = (const float*)d_in[6];
    const float* W2r = (const float*)d_in[7];

    const int N = in_sizes[0] / 256;   // 100000
    const int E = in_sizes[1] / 2;     // 320000
    const int* src = ei;
    const int* dst = ei + E;

    // Workspace layout (floats): y | h | z | aggs | agg2 | deg
    float* ws   = (float*)d_ws;
    float* y    = ws;                                  // N*128
    float* h    = y    + (size_t)N * 128;              // N*64
    float* z    = h    + (size_t)N * 64;               // N*8
    float* aggs = z    + (size_t)N * 8;                // N*64  (zeroed)
    float* agg2 = aggs + (size_t)N * 64;               // N*4   (zeroed)
    float* deg  = agg2 + (size_t)N * 4;                // N     (zeroed)
    const size_t zcount = (size_t)N * 64 + (size_t)N * 4 + (size_t)N;

    sage_zero<<<2048, 256, 0, stream>>>(aggs, zcount);

    const int mtiles = (N + 15) / 16;
    sage_gemm1<<<mtiles, 256, 0, stream>>>(x, W1l, W1r, y, N);

    sage_scatter1<<<(E * 64 + 255) / 256, 256, 0, stream>>>(y, src, dst, aggs, deg, E);

    sage_combine1<<<(N + 7) / 8, 256, 0, stream>>>(aggs, deg, b1l, y, h, N);

    sage_gemm2<<<(mtiles + 7) / 8, 256, 0, stream>>>(h, W2l, W2r, z, N);

    sage_scatter2<<<(E * 4 + 255) / 256, 256, 0, stream>>>(z, src, dst, agg2, E);

    sage_combine2<<<(N + 255) / 256, 256, 0, stream>>>(agg2, deg, b2l, z, (float*)d_out, N);
}